// DetectionConfidenceMap2keypoint_74079595922146
// MI455X (gfx1250) — compile-verified
//
#include <hip/hip_runtime.h>

typedef __attribute__((ext_vector_type(2))) float v2f;
typedef __attribute__((ext_vector_type(8))) float v8f;

#define Bn 32
#define Kn 64
#define Hn 96
#define Wn 128
#define LDS_STRIDE 132                 // 128 + 4 pad: bank-conflict-free ds_load_b64 in WMMA phase
#define MAP_ELEMS (Bn*Kn*Hn*Wn)        // 25,165,824
#define KP_OFF    MAP_ELEMS            // keypoint [B,K,2] after the map
#define ZETA_OFF  (MAP_ELEMS + Bn*Kn*2)
#define BK        (Bn*Kn)              // 2048
#define WS_PART_FLOATS (Bn*Hn*Kn*3)    // per-(b,h) partials: 589,824 floats
#define WS_OFF2   WS_PART_FLOATS       // then Sx[2048], Sy[2048], zeta[2048]

// ---------------------------------------------------------------------------
// Kernel 1: fused channel-softmax + WMMA spatial reduction.
// Grid: (Hn, Bn) blocks of 128 threads (4 waves). Thread t = column (b, h, w=t).
// ---------------------------------------------------------------------------
__global__ __launch_bounds__(128) void softmax_wmma_kernel(
    const float* __restrict__ x, float* __restrict__ out, float* __restrict__ ws)
{
    __shared__ float tile[Kn * LDS_STRIDE];   // 64 x (128+4) f32 = 33,792 B

    const int h   = blockIdx.x;
    const int b   = blockIdx.y;
    const int tid = threadIdx.x;              // == w coordinate
    const size_t colBase = ((size_t)b * Kn) * (Hn * Wn) + (size_t)h * Wn + tid;

    // ---- Phase A: softmax over the 64 channels held in registers ----
    float v[Kn];
    const float* xp = x + colBase;
    #pragma unroll
    for (int k = 0; k < Kn; ++k)
        v[k] = __builtin_nontemporal_load(xp + (size_t)k * (Hn * Wn));

    float mx = v[0];
    #pragma unroll
    for (int k = 1; k < Kn; ++k) mx = fmaxf(mx, v[k]);

    float s = 0.f;
    #pragma unroll
    for (int k = 0; k < Kn; ++k) { v[k] = __expf(v[k] - mx); s += v[k]; }
    const float r = 1.0f / s;

    float* op = out + colBase;
    #pragma unroll
    for (int k = 0; k < Kn; ++k) {
        const float p = v[k] * r;
        __builtin_nontemporal_store(p, op + (size_t)k * (Hn * Wn));
        tile[k * LDS_STRIDE + tid] = p;       // stage for matrix-engine reduction
    }
    __syncthreads();

    // ---- Phase B: D[3 x 16k] += Wt[3 x 4pos] * Map[4pos x 16k] via WMMA ----
    // Wave wv reduces channel tile kbase = 16*wv over all 128 positions (w).
    // A (documented 16x4 layout): lane L: row m=L&15, K pair {2*hi, 2*hi+1} in
    //   {vgpr0, vgpr1}; weight rows: 0 -> 1.0, 1 -> w-coord, 2 -> h-coord.
    //   Formed per chunk as one v_fma_f32 per component: a = wconst*sel1 + cBase.
    // B (4x16): lane L holds column n=L&15, rows {2*hi,2*hi+1} = two adjacent
    //   positions -> single ds_load_b64 (pairs merge into ds_load_2addr_b64).
    const int wv    = tid >> 5;
    const int ln    = tid & 31;
    const int m     = ln & 15;
    const int hi    = ln >> 4;
    const int kbase = wv * 16;

    const float sel1  = (m == 1) ? 1.0f : 0.0f;
    const float cBase = ((m == 0) ? 1.0f : ((m == 2) ? (float)h : 0.0f))
                      + sel1 * (float)(2 * hi);

    const float* trow = &tile[(kbase + m) * LDS_STRIDE + 2 * hi];
    v8f acc0 = {}, acc1 = {};                  // dual accumulators: 2x WMMA ILP
    #pragma unroll
    for (int chunk = 0; chunk < 32; chunk += 2) {
        v2f b0 = *(const v2f*)(trow + chunk * 4);
        v2f b1 = *(const v2f*)(trow + chunk * 4 + 4);
        v2f a0, a1;
        a0.x = fmaf((float)(chunk * 4    ), sel1, cBase);
        a0.y = fmaf((float)(chunk * 4 + 1), sel1, cBase);
        a1.x = fmaf((float)(chunk * 4 + 4), sel1, cBase);
        a1.y = fmaf((float)(chunk * 4 + 5), sel1, cBase);
        acc0 = __builtin_amdgcn_wmma_f32_16x16x4_f32(
                   false, a0, false, b0, (short)0, acc0, false, false);
        acc1 = __builtin_amdgcn_wmma_f32_16x16x4_f32(
                   false, a1, false, b1, (short)0, acc1, false, false);
    }

    // D rows 0/1/2 (lanes 0-15, vgprs 0-2) = zeta / Sx / Sy partials for k=kbase+n.
    if (ln < 16) {
        float* wp = ws + ((size_t)(b * Hn + h)) * (Kn * 3) + (size_t)(kbase + ln) * 3;
        wp[0] = acc0[0] + acc1[0];   // zeta partial
        wp[1] = acc0[1] + acc1[1];   // sum(map * w)
        wp[2] = acc0[2] + acc1[2];   // sum(map * h)
    }
}

// ---------------------------------------------------------------------------
// Kernel 2: deterministic fixed-order reduction over the 96 h-partials.
// 2048 threads, one per (b,k).
// ---------------------------------------------------------------------------
__global__ __launch_bounds__(256) void reduce_kernel(
    const float* __restrict__ ws, float* __restrict__ out, float* __restrict__ ws2)
{
    const int id = blockIdx.x * 256 + threadIdx.x;   // 0..2047
    const int b = id >> 6, k = id & 63;
    float z = 0.f, sx = 0.f, sy = 0.f;
    for (int h = 0; h < Hn; ++h) {
        const float* p = ws + ((size_t)(b * Hn + h)) * (Kn * 3) + (size_t)k * 3;
        z += p[0]; sx += p[1]; sy += p[2];
    }
    out[ZETA_OFF + id] = z;
    ws2[id]            = sx;
    ws2[BK + id]       = sy;
    ws2[2 * BK + id]   = z;
}

// ---------------------------------------------------------------------------
// Kernel 3: 2048-wide inclusive cumsum in flat (b,k) order + round/clamp.
// One wave: lane L owns flat indices [64L, 64L+64).
// ---------------------------------------------------------------------------
__global__ __launch_bounds__(32) void keypoint_kernel(
    const float* __restrict__ ws2, float* __restrict__ out)
{
    __shared__ float stx[32], sty[32];
    const int L = threadIdx.x;
    const float* sx = ws2;
    const float* sy = ws2 + BK;
    const float* zt = ws2 + 2 * BK;
    const int base = L * 64;

    float tx = 0.f, ty = 0.f;
    for (int i = 0; i < 64; ++i) { tx += sx[base + i]; ty += sy[base + i]; }
    stx[L] = tx; sty[L] = ty;
    __syncthreads();

    float rx = 0.f, ry = 0.f;                 // exclusive prefix of lane totals
    for (int i = 0; i < L; ++i) { rx += stx[i]; ry += sty[i]; }

    for (int i = 0; i < 64; ++i) {
        const int id = base + i;
        rx += sx[id]; ry += sy[id];
        const float z = zt[id];
        float kx = rintf(rx / z);             // jnp.round == round-half-to-even
        float ky = rintf(ry / z);
        kx = (kx > (float)Wn || kx < 0.f) ? (float)Wn * 0.5f : kx;
        ky = (ky > (float)Hn || ky < 0.f) ? (float)Hn * 0.5f : ky;
        out[KP_OFF + id * 2 + 0] = kx;
        out[KP_OFF + id * 2 + 1] = ky;
    }
}

extern "C" void kernel_launch(void* const* d_in, const int* in_sizes, int n_in,
                              void* d_out, int out_size, void* d_ws, size_t ws_size,
                              hipStream_t stream) {
    const float* x = (const float*)d_in[0];
    float* out = (float*)d_out;
    float* ws  = (float*)d_ws;       // needs (WS_PART_FLOATS + 3*BK)*4 ≈ 2.4 MB

    dim3 g1(Hn, Bn);
    softmax_wmma_kernel<<<g1, 128, 0, stream>>>(x, out, ws);
    reduce_kernel<<<BK / 256, 256, 0, stream>>>(ws, out, ws + WS_OFF2);
    keypoint_kernel<<<1, 32, 0, stream>>>(ws + WS_OFF2, out);
}